// DistanceMapPenalizedCrossEntropy_82746839924757
// MI455X (gfx1250) — compile-verified
//
#include <hip/hip_runtime.h>
#include <math.h>
#include <stdint.h>

// ---------------------------------------------------------------------------
// DistanceMapPenalizedCrossEntropy for MI455X (gfx1250, wave32)
//
// Pipeline (all on `stream`, scratch in d_ws):
//   K1 k_rowdist : per-row 1D distance-to-zero, store g^2          [8,256,256]
//   K2 k_minplus : d2 = min_k g2[k,w] + (i-k)^2; g2 tile staged in LDS via
//                  TENSOR_LOAD_TO_LDS (TDM) + s_wait_tensorcnt; then sqrt
//   K3 k_immax   : per-image max of dist                           [8]
//   K4 k_loss    : sum of (1 + dist/dmax) * bce(pred,target) via
//                  chained V_WMMA_F32_16X16X4_F32 (B = ones) reduction
//   K5 k_final   : deterministic serial sum of 64 block partials -> mean
// ---------------------------------------------------------------------------

typedef float v2f __attribute__((ext_vector_type(2)));
typedef float v8f __attribute__((ext_vector_type(8)));
typedef __attribute__((ext_vector_type(4))) unsigned int v4u;
typedef __attribute__((ext_vector_type(4))) int          v4i;
typedef __attribute__((ext_vector_type(8))) int          v8i;

#define B_IMGS 8
#define HH 256
#define WW 256
#define IMG (HH * WW)            // 65536 = 2^16
#define N_TOTAL (B_IMGS * IMG)   // 524288
#define INF_D 10000.0f           // matches reference INF = 1e4

#ifndef __has_builtin
#define __has_builtin(x) 0
#endif

#if defined(__AMDGCN__) && __has_builtin(__builtin_amdgcn_wmma_f32_16x16x4_f32)
#define USE_WMMA_REDUCE 1
#else
#define USE_WMMA_REDUCE 0
#endif

#if defined(__AMDGCN__) && __has_builtin(__builtin_amdgcn_tensor_load_to_lds) && \
    __has_builtin(__builtin_amdgcn_s_wait_tensorcnt)
#define USE_TDM 1
#else
#define USE_TDM 0
#endif

// ---------------------------------------------------------------------------
// K1: one block per image row (b*256 + h). Row distance to nearest zero pixel
// (1e4 if the row has no zero) — matches the fwd/bwd cummax construction of
// the reference exactly. Store g^2 (exact integers in fp32, or 1e8).
// ---------------------------------------------------------------------------
__global__ void k_rowdist(const float* __restrict__ target, float* __restrict__ g2) {
  const int row = blockIdx.x;   // b*256 + h
  const int w   = threadIdx.x;  // 0..255
  __shared__ float rowv[WW];
  rowv[w] = target[(size_t)row * WW + w];
  __syncthreads();
  float best = INF_D;
  #pragma unroll 8
  for (int j = 0; j < WW; ++j) {
    float d = (rowv[j] == 0.0f) ? fabsf((float)(w - j)) : INF_D;
    best = fminf(best, d);
  }
  g2[(size_t)row * WW + w] = best * best;
}

// ---------------------------------------------------------------------------
// K2: min-plus column transform. Block = (image b, 32-wide w-tile).
// The 256x32 f32 g2 tile (32 KB << 320 KB WGP LDS) is staged by the Tensor
// Data Mover: a single TENSOR_LOAD_TO_LDS with a 2-D descriptor
// (tile_dim0=32, tile_dim1=256, tensor_dim0_stride=256, data_size=4B),
// tracked by TENSORcnt. Each thread then owns one w and 32 values of i,
// reading each g2[k] from LDS once, fma+min against 32 register accumulators.
// fp32 here is exact (integers < 2^24) -> sqrt matches the reference.
// ---------------------------------------------------------------------------
__global__ void k_minplus(const float* __restrict__ g2, float* __restrict__ dist) {
  const int b   = blockIdx.x >> 3;
  const int w0  = (blockIdx.x & 7) * 32;
  const int tid = threadIdx.x;
  __shared__ float tile[HH * 32];  // tile[k*32 + wl]

  const float* gbase = g2 + (size_t)b * IMG + w0;   // tile start in global

#if USE_TDM
  if (tid == 0) {
    // --- Tensor DMA Descriptor (ISA cdna5 §8) ---
    const unsigned long long ga = (unsigned long long)(uintptr_t)gbase; // byte addr
    const unsigned int lds_off  = (unsigned int)(uintptr_t)(&tile[0]);  // LDS byte offset
    v4u g0;
    g0.x = 1u;                                    // [1:0] count=1, user D#, no gather
    g0.y = lds_off;                               // [63:32]  lds_addr
    g0.z = (unsigned int)ga;                      // [95:64]  global_addr[31:0]
    g0.w = ((unsigned int)(ga >> 32) & 0x01FFFFFFu) | (2u << 30); // addr[56:32] | type=2
    v8i g1;
    g1[0] = 0x00020000;          // workgroup_mask=0, data_size=2 (4B), no flags
    g1[1] = (int)(256u << 16);   // [63:48]  tensor_dim0 lo16 = 256
    g1[2] = (int)(256u << 16);   // [79:64] dim0 hi=0 | [95:80] tensor_dim1 lo16 = 256
    g1[3] = (int)(32u  << 16);   // [111:96] dim1 hi=0 | [127:112] tile_dim0 = 32
    g1[4] = 256;                 // [143:128] tile_dim1 = 256 | [159:144] tile_dim2 = 0
    g1[5] = 256;                 // [191:160] tensor_dim0_stride lo32 = 256
    g1[6] = 0;                   // stride0 hi16 = 0 | tensor_dim1_stride lo16 = 0
    g1[7] = 0;
    v4i z4 = {0, 0, 0, 0};       // groups 2/3: zero (2-D tile; tile_dim3/4 unused)
#if __clang_major__ >= 23
    v8i z8 = {0, 0, 0, 0, 0, 0, 0, 0};
    __builtin_amdgcn_tensor_load_to_lds(g0, g1, z4, z4, z8, 0);
#else
    __builtin_amdgcn_tensor_load_to_lds(g0, g1, z4, z4, 0);
#endif
    __builtin_amdgcn_s_wait_tensorcnt(0);         // TDM done -> tile resident in LDS
  }
  __syncthreads();
#else
  #pragma unroll
  for (int s = 0; s < 32; ++s) {
    int l = tid + 256 * s;         // 0..8191
    int k = l >> 5, wl = l & 31;
    tile[l] = gbase[(size_t)k * WW + wl];
  }
  __syncthreads();
#endif

  const int wl = tid & 31;
  const int ib = tid >> 5;         // i = ib + 8*m, m = 0..31
  float acc[32];
  #pragma unroll
  for (int m = 0; m < 32; ++m) acc[m] = 3.0e38f;

  for (int k = 0; k < HH; ++k) {
    const float g  = tile[k * 32 + wl];
    const float d0 = (float)(ib - k);
    #pragma unroll
    for (int m = 0; m < 32; ++m) {
      const float di = d0 + (float)(8 * m);
      acc[m] = fminf(acc[m], fmaf(di, di, g));
    }
  }

  float* dbase = dist + (size_t)b * IMG + w0;
  #pragma unroll
  for (int m = 0; m < 32; ++m) {
    const int i = ib + 8 * m;
    dbase[(size_t)i * WW + wl] = sqrtf(acc[m]);
  }
}

// ---------------------------------------------------------------------------
// K3: per-image max of dist (order-independent, deterministic).
// ---------------------------------------------------------------------------
__global__ void k_immax(const float* __restrict__ dist, float* __restrict__ dmax) {
  const int b   = blockIdx.x;
  const int tid = threadIdx.x;
  __shared__ float sm[256];
  const float* d = dist + (size_t)b * IMG;
  float m = 0.0f;
  for (int i = tid; i < IMG; i += 256) {
    __builtin_prefetch(d + i + 4096, 0, 0);   // -> global_prefetch_b8
    m = fmaxf(m, d[i]);
  }
  sm[tid] = m;
  __syncthreads();
  for (int s = 128; s > 0; s >>= 1) {
    if (tid < s) sm[tid] = fmaxf(sm[tid], sm[tid + s]);
    __syncthreads();
  }
  if (tid == 0) dmax[b] = sm[0];
}

// ---------------------------------------------------------------------------
// K4: weighted BCE + global reduction via the matrix engine.
// ---------------------------------------------------------------------------
__device__ __forceinline__ float elem_val(size_t ei,
                                          const float* __restrict__ pred,
                                          const float* __restrict__ target,
                                          const float* __restrict__ dist,
                                          const float* __restrict__ dmax) {
  const int   b   = (int)(ei >> 16);  // IMG == 2^16
  const float p   = pred[ei];
  const float t   = target[ei];
  const float wgt = 1.0f + dist[ei] / (dmax[b] + 1e-7f);  // ALPHA = 1
  const float bce = fmaxf(p, 0.0f) - p * t + log1pf(expf(-fabsf(p)));
  return wgt * bce;
}

__global__ void k_loss(const float* __restrict__ pred, const float* __restrict__ target,
                       const float* __restrict__ dist, const float* __restrict__ dmax,
                       float* __restrict__ partial) {
  const int tid   = threadIdx.x;
  const int wave  = tid >> 5;                 // 0..7 (wave32)
  const int lane  = tid & 31;
  const int gwave = blockIdx.x * 8 + wave;    // 0..511; 1024 elems per wave
  __shared__ float wsum[8];

  float s;
#if USE_WMMA_REDUCE
  // Matrix-engine reduction: D = A x ones(4x16) + C sums 64 fp32 values per
  // wave-step into the fp32 accumulator tile. Row sums are replicated across
  // the 16 columns, hence the final /16. EXEC stays all-ones (uniform loop).
  v8f c = {0.f, 0.f, 0.f, 0.f, 0.f, 0.f, 0.f, 0.f};
  v2f ones; ones.x = 1.0f; ones.y = 1.0f;    // B (4x16) = all ones
  for (int it = 0; it < 16; ++it) {
    const size_t e = ((size_t)gwave * 16 + it) * 64 + (size_t)lane * 2;
    v2f a;
    a.x = elem_val(e,     pred, target, dist, dmax);
    a.y = elem_val(e + 1, pred, target, dist, dmax);
    c = __builtin_amdgcn_wmma_f32_16x16x4_f32(
        /*neg_a=*/false, a, /*neg_b=*/false, ones,
        /*c_mod=*/(short)0, c, /*reuse_a=*/false, /*reuse_b=*/false);
  }
  s = 0.0f;
  #pragma unroll
  for (int r = 0; r < 8; ++r) s += c[r];
  s *= (1.0f / 16.0f);
#else
  s = 0.0f;
  for (int it = 0; it < 16; ++it) {
    const size_t e = ((size_t)gwave * 16 + it) * 64 + (size_t)lane * 2;
    s += elem_val(e, pred, target, dist, dmax);
    s += elem_val(e + 1, pred, target, dist, dmax);
  }
#endif

  // deterministic wave32 tree reduce, then across the 8 waves
  for (int off = 16; off >= 1; off >>= 1) s += __shfl_down(s, off, 32);
  if (lane == 0) wsum[wave] = s;
  __syncthreads();
  if (tid == 0) {
    float t = 0.0f;
    for (int i = 0; i < 8; ++i) t += wsum[i];
    partial[blockIdx.x] = t;
  }
}

// ---------------------------------------------------------------------------
// K5: deterministic final sum of the 64 block partials -> mean.
// ---------------------------------------------------------------------------
__global__ void k_final(const float* __restrict__ partial, float* __restrict__ out) {
  if (threadIdx.x == 0) {
    float s = 0.0f;
    for (int i = 0; i < 64; ++i) s += partial[i];
    out[0] = s * (1.0f / (float)N_TOTAL);
  }
}

// ---------------------------------------------------------------------------
extern "C" void kernel_launch(void* const* d_in, const int* in_sizes, int n_in,
                              void* d_out, int out_size, void* d_ws, size_t ws_size,
                              hipStream_t stream) {
  const float* pred   = (const float*)d_in[0];  // [8,1,256,256] f32
  const float* target = (const float*)d_in[1];  // [8,1,256,256] f32

  float* ws      = (float*)d_ws;
  float* g2      = ws;                 // 524288 f32
  float* dist    = ws + N_TOTAL;       // 524288 f32
  float* dmax    = ws + 2 * N_TOTAL;   // 8 f32
  float* partial = dmax + 8;           // 64 f32
  (void)in_sizes; (void)n_in; (void)out_size; (void)ws_size;

  k_rowdist<<<B_IMGS * HH, 256, 0, stream>>>(target, g2);
  k_minplus<<<B_IMGS * (WW / 32), 256, 0, stream>>>(g2, dist);
  k_immax  <<<B_IMGS, 256, 0, stream>>>(dist, dmax);
  k_loss   <<<64, 256, 0, stream>>>(pred, target, dist, dmax, partial);
  k_final  <<<1, 64, 0, stream>>>(partial, (float*)d_out);
}